// GraphAttentionConvLayer_21071109554804
// MI455X (gfx1250) — compile-verified
//
#include <hip/hip_runtime.h>

#define GAT_ALPHA 0.2f
#define N_NODES   50000
#define N_EDGES   800000
#define DIM       128

typedef __attribute__((ext_vector_type(2))) float v2f;
typedef __attribute__((ext_vector_type(8))) float v8f;

// Padded LDS pitch (floats) for the transposed weight tile:
// pitch 132 -> bank(4n+k), conflict-free ds_load_b64 across a wave, 8B aligned.
#define WT_PITCH 132

// ---------------------------------------------------------------------------
// Kernel 1: H = X @ W + bias  via V_WMMA_F32_16X16X4_F32 (full f32 precision).
// 256 threads = 8 waves; each wave computes a 16-row x 128-col tile.
// ---------------------------------------------------------------------------
__global__ __launch_bounds__(256) void gat_gemm_wmma(
    const float* __restrict__ X,     // [N_NODES][DIM]
    const float* __restrict__ W,     // [DIM][DIM]
    const float* __restrict__ bias,  // [DIM]
    float* __restrict__ H)           // [N_NODES][DIM]
{
  __shared__ float Wt[DIM * WT_PITCH];  // Wt[n*WT_PITCH + k] = W[k][n]

  const int tid = threadIdx.x;
  // Cooperative transpose of W into LDS (one-time cost per block).
  for (int idx = tid; idx < DIM * DIM; idx += 256) {
    const int k = idx >> 7;
    const int n = idx & (DIM - 1);
    Wt[n * WT_PITCH + k] = W[idx];
  }
  __syncthreads();

  const int wave = tid >> 5;
  const int lane = tid & 31;
  const int l16  = lane & 15;
  const int hi   = lane >> 4;                       // 0 or 1
  const int rowBase = (blockIdx.x * 8 + wave) * 16; // 16 rows per wave

  // A-fragment row for this lane (ISA 16x4 f32 layout: lanes hold M=l16,
  // K = 2*hi + {0,1}); clamp for the ragged tail so EXEC stays all-1s.
  int arow = rowBase + l16;
  if (arow >= N_NODES) arow = N_NODES - 1;
  const float* aptr = X + (size_t)arow * DIM + 2 * hi;

  // C/D layout: lane owns column n = 16*t + l16; VGPR r holds row 8*hi + r.
  v8f acc[8];
  #pragma unroll
  for (int t = 0; t < 8; ++t) {
    const float b = bias[16 * t + l16];
    acc[t] = (v8f){b, b, b, b, b, b, b, b};
  }

  for (int k0 = 0; k0 < DIM; k0 += 4) {
    const v2f afrag = *(const v2f*)(aptr + k0);
    #pragma unroll
    for (int t = 0; t < 8; ++t) {
      const int n = 16 * t + l16;
      // B-fragment (4x16, KxN): lane holds column n, K = 2*hi + {0,1}.
      const v2f bfrag = *(const v2f*)(&Wt[n * WT_PITCH + k0 + 2 * hi]);
      acc[t] = __builtin_amdgcn_wmma_f32_16x16x4_f32(
          /*neg_a=*/false, afrag, /*neg_b=*/false, bfrag,
          /*c_mod=*/(short)0, acc[t], /*reuse_a=*/false, /*reuse_b=*/false);
    }
  }

  #pragma unroll
  for (int r = 0; r < 8; ++r) {
    const int m = rowBase + 8 * hi + r;
    if (m < N_NODES) {
      #pragma unroll
      for (int t = 0; t < 8; ++t)
        H[(size_t)m * DIM + 16 * t + l16] = acc[t][r];
    }
  }
}

// ---------------------------------------------------------------------------
// Kernel 2: per-node attention logits. One wave32 per node; float4 per lane;
// butterfly reduce across the wave.
// ---------------------------------------------------------------------------
__global__ __launch_bounds__(256) void gat_scores(
    const float* __restrict__ H, const float* __restrict__ a,
    float* __restrict__ s_src, float* __restrict__ s_dst)
{
  const int node = blockIdx.x * 8 + (threadIdx.x >> 5);
  const int lane = threadIdx.x & 31;
  if (node >= N_NODES) return;

  const float4 hv = ((const float4*)(H + (size_t)node * DIM))[lane];
  const float4 as = ((const float4*)a)[lane];
  const float4 ad = ((const float4*)(a + DIM))[lane];

  float ps = hv.x * as.x + hv.y * as.y + hv.z * as.z + hv.w * as.w;
  float pd = hv.x * ad.x + hv.y * ad.y + hv.z * ad.z + hv.w * ad.w;
  #pragma unroll
  for (int off = 16; off > 0; off >>= 1) {
    ps += __shfl_xor(ps, off, 32);
    pd += __shfl_xor(pd, off, 32);
  }
  if (lane == 0) { s_src[node] = ps; s_dst[node] = pd; }
}

// ---------------------------------------------------------------------------
// Kernel 3: edge scatter. One wave32 per edge; lane owns a float4 slice.
// h/out live in L2 (192 MB), so this is L2 atomic throughput bound; use
// hardware global_atomic_add_f32 via unsafeAtomicAdd.
// ---------------------------------------------------------------------------
__global__ __launch_bounds__(256) void gat_scatter(
    const float* __restrict__ H,
    const int* __restrict__ srcIdx, const int* __restrict__ dstIdx,
    const float* __restrict__ s_src, const float* __restrict__ s_dst,
    float* __restrict__ out)
{
  const int edge = blockIdx.x * 8 + (threadIdx.x >> 5);
  const int lane = threadIdx.x & 31;
  if (edge >= N_EDGES) return;

  const int s = srcIdx[edge];
  const int d = dstIdx[edge];
  float e = s_src[s] + s_dst[d];
  e = (e > 0.0f) ? e : GAT_ALPHA * e;

  const float4 hv = ((const float4*)(H + (size_t)d * DIM))[lane];
  float* o = out + (size_t)s * DIM + 4 * lane;
  unsafeAtomicAdd(o + 0, e * hv.x);
  unsafeAtomicAdd(o + 1, e * hv.y);
  unsafeAtomicAdd(o + 2, e * hv.z);
  unsafeAtomicAdd(o + 3, e * hv.w);
}

// ---------------------------------------------------------------------------
extern "C" void kernel_launch(void* const* d_in, const int* in_sizes, int n_in,
                              void* d_out, int out_size, void* d_ws, size_t ws_size,
                              hipStream_t stream) {
  const float* features = (const float*)d_in[0];  // [50000][128]
  const float* weight   = (const float*)d_in[1];  // [128][128]
  const float* a        = (const float*)d_in[2];  // [256]
  const float* bias     = (const float*)d_in[3];  // [128]
  const int*   adj      = (const int*)d_in[4];    // [2][800000]

  float* out   = (float*)d_out;
  float* H     = (float*)d_ws;                       // 25.6 MB
  float* s_src = H + (size_t)N_NODES * DIM;          // 200 KB
  float* s_dst = s_src + N_NODES;                    // 200 KB

  // Zero output before atomic accumulation (graph-capture safe).
  hipMemsetAsync(d_out, 0, (size_t)N_NODES * DIM * sizeof(float), stream);

  gat_gemm_wmma<<<(N_NODES + 127) / 128, 256, 0, stream>>>(features, weight, bias, H);
  gat_scores<<<(N_NODES + 7) / 8, 256, 0, stream>>>(H, a, s_src, s_dst);
  gat_scatter<<<(N_EDGES + 7) / 8, 256, 0, stream>>>(H, adj, adj + N_EDGES,
                                                     s_src, s_dst, out);
}